// Tokenizer_27565100106296
// MI455X (gfx1250) — compile-verified
//
#include <hip/hip_runtime.h>
#include <math.h>

typedef float v2f __attribute__((ext_vector_type(2)));
typedef float v8f __attribute__((ext_vector_type(8)));

#define KDIM 256      // C
#define VDIM 1024     // V (codebook rows used)
#define ROWS 16       // rows per workgroup
#define TDIM 2048     // T (sequence length)

// ws layout (floats): [0]=commit_sum [1]=smooth_sum [2]=mask_sum [3..3+VDIM)=counts
#define WS_FLOATS (3 + VDIM)

__global__ void vq_init(float* __restrict__ ws) {
    int t = blockIdx.x * blockDim.x + threadIdx.x;
    if (t < WS_FLOATS) ws[t] = 0.0f;
}

__global__ __launch_bounds__(256) void vq_main(
        const float* __restrict__ z,      // [N, 256]
        const float* __restrict__ mask,   // [N]
        const float* __restrict__ cw,     // [1025, 256]
        float* __restrict__ out_logp,     // [N, 1024]
        float* __restrict__ out_zq,       // [N, 256]
        float* __restrict__ ws)
{
    __shared__ float lds_cos[ROWS * VDIM];   // 64 KB f32 tile

    const float* e = cw + KDIM;              // codebook_weight[1:, :]
    const int tid  = threadIdx.x;
    const int wave = tid >> 5;
    const int lane = tid & 31;
    const int half = lane >> 4;              // 0: K=0,1  1: K=2,3
    const int l16  = lane & 15;
    const int row0 = blockIdx.x * ROWS;

    // ---------------- GEMM: rows [row0,row0+16) x cols [wave*128, +128) ----------------
    const int colbase = wave * 128;
    v8f acc[8];
    const v8f vzero = {0.f,0.f,0.f,0.f,0.f,0.f,0.f,0.f};
#pragma unroll
    for (int t = 0; t < 8; ++t) acc[t] = vzero;

    // A fragment (16x4 f32): lane l16 -> row, half selects K pair
    const float* pa = z + (size_t)(row0 + l16) * KDIM + 2 * half;
    // B fragment (4x16 f32): lane l16 -> column, half selects K pair
    const float* pb = e + (size_t)(colbase + l16) * KDIM + 2 * half;

    for (int k = 0; k < KDIM; k += 4) {
        v2f a = *(const v2f*)(pa + k);
#pragma unroll
        for (int t = 0; t < 8; ++t) {
            v2f b = *(const v2f*)(pb + k + t * (16 * KDIM));
            acc[t] = __builtin_amdgcn_wmma_f32_16x16x4_f32(
                false, a, false, b, (short)0, acc[t], false, false);
        }
    }

    // C/D layout: vgpr r holds row r (lanes 0-15) / row r+8 (lanes 16-31), col = l16
#pragma unroll
    for (int t = 0; t < 8; ++t) {
        const int col = colbase + t * 16 + l16;
#pragma unroll
        for (int r = 0; r < 8; ++r) {
            lds_cos[(r + 8 * half) * VDIM + col] = acc[t][r];
        }
    }
    __syncthreads();

    // ---------------- per-row softmax / argmax (16-thread group per row) ----------------
    const int row = tid >> 4;                // 0..15
    const int s   = tid & 15;                // lane-in-group; group = one half-wave
    const float* rowp = lds_cos + row * VDIM;

    // max + first-index argmax over this row (strided scan keeps LDS banks spread)
    float m = -INFINITY;
    int   bi = VDIM;
    for (int j = 0; j < VDIM / 16; ++j) {
        const int c = j * 16 + s;
        const float v = rowp[c];
        if (v > m) { m = v; bi = c; }
    }
#pragma unroll
    for (int d = 1; d < 16; d <<= 1) {       // xor 1/2/4/8 stays inside the half-wave
        float om = __shfl_xor(m,  d, 32);
        int  obi = __shfl_xor(bi, d, 32);
        if (om > m || (om == m && obi < bi)) { m = om; bi = obi; }
    }

    // sum of exp(x - m)
    float se = 0.0f;
    for (int j = 0; j < VDIM / 16; ++j) se += expf(rowp[j * 16 + s] - m);
#pragma unroll
    for (int d = 1; d < 16; d <<= 1) se += __shfl_xor(se, d, 32);
    const float lse = m + logf(se);

    // log_probs = x - lse  (16 consecutive floats per group per step)
    float* outp = out_logp + (size_t)(row0 + row) * VDIM;
    for (int j = 0; j < VDIM / 16; ++j) {
        const int c = j * 16 + s;
        outp[c] = rowp[c] - lse;
    }

    // ---------------- gather z_q, commitment & smoothness partials ----------------
    const size_t g    = (size_t)row0 + row;
    const float  mrow = mask[g];
    const float* erow = e + (size_t)bi * KDIM;
    const float* zrow = z + g * KDIM;
    float* zqrow      = out_zq + g * KDIM;

    float csum = 0.0f;
    for (int j = 0; j < KDIM / 16; ++j) {
        const int c = j * 16 + s;
        const float ev = erow[c];
        zqrow[c] = ev;
        const float d = zrow[c] - ev;
        csum += d * d;
    }
    csum *= mrow;

    float ssm = 0.0f;
    if ((g % TDIM) != (TDIM - 1)) {          // term (z[t] - z[t+1])^2 * mask[t+1]
        const float mnext = mask[g + 1];
        const float* znext = zrow + KDIM;
        float acc2 = 0.0f;
        for (int j = 0; j < KDIM / 16; ++j) {
            const int c = j * 16 + s;
            const float d = zrow[c] - znext[c];
            acc2 += d * d;
        }
        ssm = acc2 * mnext;
    }
#pragma unroll
    for (int d = 1; d < 16; d <<= 1) {
        csum += __shfl_xor(csum, d, 32);
        ssm  += __shfl_xor(ssm,  d, 32);
    }
    if (s == 0) {
        atomicAdd(&ws[0], csum);
        atomicAdd(&ws[1], ssm);
        atomicAdd(&ws[2], mrow);
        atomicAdd(&ws[3 + bi], mrow);        // codebook usage counts
    }
}

__global__ void vq_finalize(const float* __restrict__ ws,
                            float* __restrict__ d_out, int N)
{
    const float msum  = ws[2];
    const float denom = msum * (float)KDIM;
    const int t = threadIdx.x;
    const size_t emoff = 2 + (size_t)N * VDIM + (size_t)N * KDIM;
    if (t < VDIM) d_out[emoff + t] = ws[3 + t] / msum;   // e_mean
    if (t == 0) {
        d_out[0] = ws[1] / denom;   // smoothness_loss
        d_out[1] = ws[0] / denom;   // commitment_loss
    }
}

extern "C" void kernel_launch(void* const* d_in, const int* in_sizes, int n_in,
                              void* d_out, int out_size, void* d_ws, size_t ws_size,
                              hipStream_t stream) {
    const float* z    = (const float*)d_in[0];     // [B,T,C]
    const float* mask = (const float*)d_in[1];     // [B,T,1]
    const float* cw   = (const float*)d_in[2];     // [1025,C]
    float* out = (float*)d_out;
    float* ws  = (float*)d_ws;

    const int N = in_sizes[0] / KDIM;              // B*T = 32768

    // out layout: [0]=smooth, [1]=commit, [2 .. 2+N*V) log_probs,
    //             [.. +N*C) z_q, [.. +V) e_mean
    float* out_logp = out + 2;
    float* out_zq   = out + 2 + (size_t)N * VDIM;

    vq_init<<<(WS_FLOATS + 255) / 256, 256, 0, stream>>>(ws);
    vq_main<<<N / ROWS, 256, 0, stream>>>(z, mask, cw, out_logp, out_zq, ws);
    vq_finalize<<<1, VDIM, 0, stream>>>(ws, out, N);
}